// Encoder_3478923510230
// MI455X (gfx1250) — compile-verified
//
#include <hip/hip_runtime.h>
#include <hip/hip_bf16.h>
#include <math.h>

// ---------------------------------------------------------------------------
// Encoder block fwd for MI455X (gfx1250, wave32, WMMA).
// All GEMMs use V_WMMA_F32_16X16X4_F32 (full fp32 precision, matrix pipe).
// Attention is flash-style with double-buffered ASYNC global->LDS K/V tiles
// (GLOBAL_LOAD_ASYNC_TO_LDS_B128 + s_wait_asynccnt, CDNA5 async path).
// ---------------------------------------------------------------------------

typedef float v2f __attribute__((ext_vector_type(2)));
typedef float v8f __attribute__((ext_vector_type(8)));

#define EMBED    384
#define NTOK     8192          // B * N = 2 * 4096
#define SEQ      4096
#define HEADS    6
#define HD       64
#define QKV_OUT  1152          // 3 * EMBED
#define HID      1536          // 4 * EMBED

__device__ __forceinline__ v8f wmma_f32(v2f a, v2f b, v8f c) {
    // D = A(16x4 f32) * B(4x16 f32) + C(16x16 f32)
    return __builtin_amdgcn_wmma_f32_16x16x4_f32(
        false, a, false, b, (short)0, c, false, false);
}

// LDS byte address (addr[31:0] of the flat LDS aperture address).
__device__ __forceinline__ unsigned lds_addr_u32(const void* p) {
    return (unsigned)(size_t)(__attribute__((address_space(3))) const char*)p;
}

// CDNA5 async copy: 16 bytes per lane, memory -> LDS, tracked by ASYNCcnt.
__device__ __forceinline__ void async_ld_b128(unsigned lds_dst, const void* g) {
    asm volatile("global_load_async_to_lds_b128 %0, %1, off"
                 :: "v"(lds_dst), "v"((unsigned long long)(size_t)g)
                 : "memory");
}

// Wave-level GEMM: one wave computes a 16(M) x 64(N) tile of C = A*B.
// A fragment (16x4 f32): lanes 0-15 hold rows M=0..15 at K=k0,k0+1;
//                        lanes 16-31 hold rows M=0..15 at K=k0+2,k0+3.
// B fragment (4x16 f32): lane l16 holds column N=l16; v0 = row k0 (+2 hi
//                        half), v1 = row k0+1 (+2 hi half).
__device__ __forceinline__ void gemm_tile_16x64(const float* __restrict__ A, int lda,
                                                const float* __restrict__ B, int ldb,
                                                int K, v8f acc[4]) {
    const int lane = threadIdx.x & 31;
    const int half = lane >> 4;
    const int l16  = lane & 15;
    const float* arow = A + (size_t)l16 * lda + 2 * half;
    for (int k0 = 0; k0 < K; k0 += 4) {
        v2f a;
        a.x = arow[k0];
        a.y = arow[k0 + 1];
        const float* brow = B + (size_t)(k0 + 2 * half) * ldb + l16;
#pragma unroll
        for (int nt = 0; nt < 4; ++nt) {
            v2f b;
            b.x = brow[nt * 16];
            b.y = brow[nt * 16 + ldb];
            acc[nt] = wmma_f32(a, b, acc[nt]);
        }
    }
}

// ---------------------------------------------------------------------------
// LayerNorm: one wave per token (384 = 32 lanes x 12 elems).
// ---------------------------------------------------------------------------
__global__ void ln_kernel(const float* __restrict__ x, const float* __restrict__ g,
                          const float* __restrict__ be, float* __restrict__ out) {
    const int token = blockIdx.x * 8 + (threadIdx.x >> 5);
    const int lane  = threadIdx.x & 31;
    const float* xp = x + (size_t)token * EMBED;
    float vals[12];
    float sum = 0.f;
#pragma unroll
    for (int i = 0; i < 12; ++i) {
        vals[i] = xp[lane + i * 32];
        sum += vals[i];
    }
#pragma unroll
    for (int off = 16; off >= 1; off >>= 1) sum += __shfl_xor(sum, off, 32);
    const float mu = sum * (1.0f / EMBED);
    float var = 0.f;
#pragma unroll
    for (int i = 0; i < 12; ++i) {
        float d = vals[i] - mu;
        var += d * d;
    }
#pragma unroll
    for (int off = 16; off >= 1; off >>= 1) var += __shfl_xor(var, off, 32);
    const float rstd = rsqrtf(var * (1.0f / EMBED) + 1e-5f);
    float* op = out + (size_t)token * EMBED;
#pragma unroll
    for (int i = 0; i < 12; ++i) {
        const int c = lane + i * 32;
        op[c] = (vals[i] - mu) * rstd * g[c] + be[c];
    }
}

// ---------------------------------------------------------------------------
// QKV GEMM: qkv = h @ w_qkv + b_qkv, scattered to (B,H,N,64) Q/K/V buffers.
// Column c in [0,1152): head = c/192, w = c%192 -> q(w<64) k(w<128) v(else).
// ---------------------------------------------------------------------------
__global__ void gemm_qkv_kernel(const float* __restrict__ h, const float* __restrict__ w,
                                const float* __restrict__ bias, float* __restrict__ qb,
                                float* __restrict__ kb, float* __restrict__ vb) {
    const int wid = blockIdx.x * 8 + (threadIdx.x >> 5);
    const int tiles_n = QKV_OUT / 64;                    // 18
    const int tm = wid / tiles_n, tn = wid % tiles_n;
    v8f acc[4] = {};
    gemm_tile_16x64(h + (size_t)tm * 16 * EMBED, EMBED, w + tn * 64, QKV_OUT,
                    EMBED, acc);
    const int lane = threadIdx.x & 31;
    const int half = lane >> 4, l16 = lane & 15;
#pragma unroll
    for (int nt = 0; nt < 4; ++nt) {
#pragma unroll
        for (int r = 0; r < 8; ++r) {
            const int row = tm * 16 + r + 8 * half;      // token
            const int col = tn * 64 + nt * 16 + l16;     // qkv column
            const float val = acc[nt][r] + bias[col];
            const int b_ = row >> 12;                    // token / 4096
            const int n_ = row & (SEQ - 1);
            const int head = col / 192;
            const int wi   = col % 192;
            const size_t base = (((size_t)b_ * HEADS + head) * SEQ + n_) * HD;
            if (wi < 64)       qb[base + wi]        = val;
            else if (wi < 128) kb[base + (wi - 64)] = val;
            else               vb[base + (wi - 128)] = val;
        }
    }
}

// ---------------------------------------------------------------------------
// Flash attention: grid (N/64, B*H); block 128 (4 waves, each a 16-query tile).
// K/V 16-key tiles are double-buffered in LDS via async global->LDS b128
// copies; the next tile streams in while the current one feeds the WMMAs.
// ---------------------------------------------------------------------------
__global__ void attn_kernel(const float* __restrict__ q, const float* __restrict__ k,
                            const float* __restrict__ v, float* __restrict__ msa) {
    __shared__ float lds_k[2][16 * HD];
    __shared__ float lds_v[2][16 * HD];
    __shared__ float lds_p[4 * 16 * 16];

    const int bh   = blockIdx.y;
    const float* qp = q + (size_t)bh * SEQ * HD;
    const float* kp = k + (size_t)bh * SEQ * HD;
    const float* vp = v + (size_t)bh * SEQ * HD;
    const int t    = threadIdx.x;
    const int wave = t >> 5;
    const int lane = t & 31;
    const int half = lane >> 4, l16 = lane & 15;
    const int q0   = blockIdx.x * 64 + wave * 16;

    // Per-thread async-copy slots: float4 elements t and t+128 of each tile.
    const unsigned kd0[2] = { lds_addr_u32(&lds_k[0][0]) + (unsigned)t * 16u,
                              lds_addr_u32(&lds_k[1][0]) + (unsigned)t * 16u };
    const unsigned vd0[2] = { lds_addr_u32(&lds_v[0][0]) + (unsigned)t * 16u,
                              lds_addr_u32(&lds_v[1][0]) + (unsigned)t * 16u };

    // Preload the wave's Q tile (16x64) as 16 A-fragments (held in regs).
    v2f qa[16];
#pragma unroll
    for (int ks = 0; ks < 16; ++ks) {
        const float* p = qp + (size_t)(q0 + l16) * HD + ks * 4 + 2 * half;
        qa[ks].x = p[0];
        qa[ks].y = p[1];
    }

    v8f o[4] = {};
    float m_run[8], l_run[8];
#pragma unroll
    for (int r = 0; r < 8; ++r) { m_run[r] = -3.0e38f; l_run[r] = 0.f; }
    const float scale = 0.125f;                           // 1/sqrt(64)
    float* pl = lds_p + wave * 256;

    // Prologue: kick off tile 0 into buffer 0 (4 async instrs per thread).
    {
        const char* kg = (const char*)kp + (size_t)t * 16u;
        const char* vg = (const char*)vp + (size_t)t * 16u;
        async_ld_b128(kd0[0],         kg);
        async_ld_b128(kd0[0] + 2048u, kg + 2048u);
        async_ld_b128(vd0[0],         vg);
        async_ld_b128(vd0[0] + 2048u, vg + 2048u);
    }

    const int ntiles = SEQ / 16;                          // 256
    for (int jt = 0; jt < ntiles; ++jt) {
        const int cur = jt & 1;
        if (jt + 1 < ntiles) {                            // prefetch next tile
            const int nxt = cur ^ 1;
            const size_t gb = (size_t)(jt + 1) * 16 * HD * sizeof(float);
            const char* kg = (const char*)kp + gb + (size_t)t * 16u;
            const char* vg = (const char*)vp + gb + (size_t)t * 16u;
            async_ld_b128(kd0[nxt],         kg);
            async_ld_b128(kd0[nxt] + 2048u, kg + 2048u);
            async_ld_b128(vd0[nxt],         vg);
            async_ld_b128(vd0[nxt] + 2048u, vg + 2048u);
            asm volatile("s_wait_asynccnt 4" ::: "memory"); // cur tile resident
        } else {
            asm volatile("s_wait_asynccnt 0" ::: "memory");
        }
        __syncthreads();                                  // all waves' loads done

        const float* lk = lds_k[cur];
        const float* lv = lds_v[cur];

        // S = Q @ K^T  (16x16); B[K=d][N=jl] = Ktile[jl][d]
        v8f s = {};
#pragma unroll
        for (int ks = 0; ks < 16; ++ks) {
            const int d0 = ks * 4 + 2 * half;
            v2f b;
            b.x = lk[l16 * HD + d0];
            b.y = lk[l16 * HD + d0 + 1];
            s = wmma_f32(qa[ks], b, s);
        }

        // Online softmax update (row r lives in lanes 0-15 / 16-31 of reg r).
#pragma unroll
        for (int r = 0; r < 8; ++r) {
            float sv = s[r] * scale;
            float mx = sv;
#pragma unroll
            for (int off = 8; off >= 1; off >>= 1) mx = fmaxf(mx, __shfl_xor(mx, off, 32));
            const float m_new = fmaxf(m_run[r], mx);
            const float p     = __expf(sv - m_new);
            float ls = p;
#pragma unroll
            for (int off = 8; off >= 1; off >>= 1) ls += __shfl_xor(ls, off, 32);
            const float corr = __expf(m_run[r] - m_new);
            l_run[r] = l_run[r] * corr + ls;
            m_run[r] = m_new;
#pragma unroll
            for (int nt = 0; nt < 4; ++nt) o[nt][r] *= corr;
            s[r] = p;
        }

        // Transpose P (C-layout -> A-layout) through per-wave LDS scratch.
        asm volatile("" ::: "memory");
#pragma unroll
        for (int r = 0; r < 8; ++r) pl[(r + 8 * half) * 16 + l16] = s[r];
        asm volatile("s_wait_dscnt 0" ::: "memory");

        // O += P @ Vtile ; B[K=jl][N=d] = Vtile[jl][d]
#pragma unroll
        for (int ks = 0; ks < 4; ++ks) {
            const int kk = ks * 4 + 2 * half;
            v2f a;
            a.x = pl[l16 * 16 + kk];
            a.y = pl[l16 * 16 + kk + 1];
#pragma unroll
            for (int nt = 0; nt < 4; ++nt) {
                v2f b;
                b.x = lv[kk * HD + nt * 16 + l16];
                b.y = lv[(kk + 1) * HD + nt * 16 + l16];
                o[nt] = wmma_f32(a, b, o[nt]);
            }
        }
        __syncthreads();   // all waves done reading cur before it is refilled
    }

    // Normalize and store to msa (token-major, col = h*64 + d).
    const int b_ = bh / HEADS, h_ = bh % HEADS;
#pragma unroll
    for (int nt = 0; nt < 4; ++nt) {
#pragma unroll
        for (int r = 0; r < 8; ++r) {
            const int row = q0 + r + 8 * half;
            const int col = h_ * HD + nt * 16 + l16;
            msa[((size_t)b_ * SEQ + row) * EMBED + col] = o[nt][r] / l_run[r];
        }
    }
}

// ---------------------------------------------------------------------------
// Projection + residual: x1 = msa @ w_lin + b_lin + x
// ---------------------------------------------------------------------------
__global__ void proj_kernel(const float* __restrict__ msa, const float* __restrict__ w,
                            const float* __restrict__ bias, const float* __restrict__ x,
                            float* __restrict__ x1) {
    const int wid = blockIdx.x * 8 + (threadIdx.x >> 5);
    const int tiles_n = EMBED / 64;                       // 6
    const int tm = wid / tiles_n, tn = wid % tiles_n;
    v8f acc[4] = {};
    gemm_tile_16x64(msa + (size_t)tm * 16 * EMBED, EMBED, w + tn * 64, EMBED,
                    EMBED, acc);
    const int lane = threadIdx.x & 31;
    const int half = lane >> 4, l16 = lane & 15;
#pragma unroll
    for (int nt = 0; nt < 4; ++nt) {
#pragma unroll
        for (int r = 0; r < 8; ++r) {
            const int row = tm * 16 + r + 8 * half;
            const int col = tn * 64 + nt * 16 + l16;
            const size_t idx = (size_t)row * EMBED + col;
            x1[idx] = acc[nt][r] + bias[col] + x[idx];
        }
    }
}

// ---------------------------------------------------------------------------
// MLP fc1 + exact GELU: hidden = gelu(h2 @ w1 + b1)
// ---------------------------------------------------------------------------
__global__ void mlp1_kernel(const float* __restrict__ h2, const float* __restrict__ w,
                            const float* __restrict__ bias, float* __restrict__ hid) {
    const int wid = blockIdx.x * 8 + (threadIdx.x >> 5);
    const int tiles_n = HID / 64;                         // 24
    const int tm = wid / tiles_n, tn = wid % tiles_n;
    v8f acc[4] = {};
    gemm_tile_16x64(h2 + (size_t)tm * 16 * EMBED, EMBED, w + tn * 64, HID,
                    EMBED, acc);
    const int lane = threadIdx.x & 31;
    const int half = lane >> 4, l16 = lane & 15;
#pragma unroll
    for (int nt = 0; nt < 4; ++nt) {
#pragma unroll
        for (int r = 0; r < 8; ++r) {
            const int row = tm * 16 + r + 8 * half;
            const int col = tn * 64 + nt * 16 + l16;
            const float vx = acc[nt][r] + bias[col];
            hid[(size_t)row * HID + col] =
                0.5f * vx * (1.0f + erff(vx * 0.70710678118654752440f));
        }
    }
}

// ---------------------------------------------------------------------------
// MLP fc2 + residual: out = hidden @ w2 + b2 + x1
// ---------------------------------------------------------------------------
__global__ void mlp2_kernel(const float* __restrict__ hid, const float* __restrict__ w,
                            const float* __restrict__ bias, const float* __restrict__ x1,
                            float* __restrict__ out) {
    const int wid = blockIdx.x * 8 + (threadIdx.x >> 5);
    const int tiles_n = EMBED / 64;                       // 6
    const int tm = wid / tiles_n, tn = wid % tiles_n;
    v8f acc[4] = {};
    gemm_tile_16x64(hid + (size_t)tm * 16 * HID, HID, w + tn * 64, EMBED,
                    HID, acc);
    const int lane = threadIdx.x & 31;
    const int half = lane >> 4, l16 = lane & 15;
#pragma unroll
    for (int nt = 0; nt < 4; ++nt) {
#pragma unroll
        for (int r = 0; r < 8; ++r) {
            const int row = tm * 16 + r + 8 * half;
            const int col = tn * 64 + nt * 16 + l16;
            const size_t idx = (size_t)row * EMBED + col;
            out[idx] = acc[nt][r] + bias[col] + x1[idx];
        }
    }
}

// ---------------------------------------------------------------------------
// Launch: ws layout (each 12 MB = 8192*384*4):
//   [h @0][q @12M][k @24M][v @36M][msa @48M][x1 @60M]; MLP hidden (48 MB)
//   overlays q..msa once they are dead. Total = 72 MB.
// ---------------------------------------------------------------------------
extern "C" void kernel_launch(void* const* d_in, const int* in_sizes, int n_in,
                              void* d_out, int out_size, void* d_ws, size_t ws_size,
                              hipStream_t stream) {
    const float* x     = (const float*)d_in[0];
    const float* w_qkv = (const float*)d_in[1];
    const float* b_qkv = (const float*)d_in[2];
    const float* w_lin = (const float*)d_in[3];
    const float* b_lin = (const float*)d_in[4];
    const float* g1    = (const float*)d_in[5];
    const float* be1   = (const float*)d_in[6];
    const float* g2    = (const float*)d_in[7];
    const float* be2   = (const float*)d_in[8];
    const float* w1    = (const float*)d_in[9];
    const float* b1    = (const float*)d_in[10];
    const float* w2    = (const float*)d_in[11];
    const float* b2    = (const float*)d_in[12];
    float* out = (float*)d_out;

    const size_t BUF = (size_t)NTOK * EMBED * sizeof(float);   // 12 MB
    char* ws = (char*)d_ws;
    float* buf_h   = (float*)(ws);
    float* buf_q   = (float*)(ws + 1 * BUF);
    float* buf_k   = (float*)(ws + 2 * BUF);
    float* buf_v   = (float*)(ws + 3 * BUF);
    float* buf_msa = (float*)(ws + 4 * BUF);
    float* buf_x1  = (float*)(ws + 5 * BUF);
    float* buf_hid = (float*)(ws + 1 * BUF);                   // overlays q..msa

    // 1. LN1
    ln_kernel<<<NTOK / 8, 256, 0, stream>>>(x, g1, be1, buf_h);
    // 2. QKV GEMM (8192x1152, K=384): 512*18 wave tiles / 8 waves per block
    gemm_qkv_kernel<<<(512 * 18) / 8, 256, 0, stream>>>(buf_h, w_qkv, b_qkv,
                                                        buf_q, buf_k, buf_v);
    // 3. Flash attention (async double-buffered K/V)
    attn_kernel<<<dim3(SEQ / 64, 2 * HEADS), 128, 0, stream>>>(buf_q, buf_k,
                                                               buf_v, buf_msa);
    // 4. Projection + residual
    proj_kernel<<<(512 * 6) / 8, 256, 0, stream>>>(buf_msa, w_lin, b_lin, x, buf_x1);
    // 5. LN2 (reuses buf_h)
    ln_kernel<<<NTOK / 8, 256, 0, stream>>>(buf_x1, g2, be2, buf_h);
    // 6. MLP fc1 + GELU (8192x1536, K=384)
    mlp1_kernel<<<(512 * 24) / 8, 256, 0, stream>>>(buf_h, w1, b1, buf_hid);
    // 7. MLP fc2 + residual (8192x384, K=1536)
    mlp2_kernel<<<(512 * 6) / 8, 256, 0, stream>>>(buf_hid, w2, b2, buf_x1, out);
}